// encoder_5849745457603
// MI455X (gfx1250) — compile-verified
//
#include <hip/hip_runtime.h>
#include <math.h>

// ---------------------------------------------------------------------------
// Types / helpers
// ---------------------------------------------------------------------------
typedef __attribute__((ext_vector_type(2))) float v2f;
typedef __attribute__((ext_vector_type(8))) float v8f;

#define WMMA_F32_K4(a, b, c) \
  __builtin_amdgcn_wmma_f32_16x16x4_f32(false, (a), false, (b), (short)0, (c), false, false)

#if defined(__AMDGCN__) && __has_builtin(__builtin_amdgcn_global_load_async_to_lds_b128)
#define USE_ASYNC_LDS 1
#else
#define USE_ASYNC_LDS 0
#endif

#if USE_ASYNC_LDS
// Builtin signature (from clang diagnostic): param 1 is
//   'int __attribute__((vector_size(16))) __device__ *'  i.e. v4i in AS(1).
typedef int v4i_g __attribute__((vector_size(16)));
typedef __attribute__((address_space(1))) v4i_g as1_v4i;
typedef __attribute__((address_space(3))) v4i_g as3_v4i;

__device__ __forceinline__ void async_copy16(const float* g, float* l) {
  // GLOBAL_LOAD_ASYNC_TO_LDS_B128: 16B per lane, tracked by ASYNCcnt.
  __builtin_amdgcn_global_load_async_to_lds_b128((as1_v4i*)g, (as3_v4i*)l, 0,
                                                 0);
}

template <int N>
__device__ __forceinline__ void wait_asynccnt() {
#if __has_builtin(__builtin_amdgcn_s_wait_asynccnt)
  __builtin_amdgcn_s_wait_asynccnt((unsigned short)N);
#else
  asm volatile("s_wait_asynccnt %0" ::"i"(N) : "memory");
#endif
}
#endif

__device__ __forceinline__ float wave_sum(float v) {
  for (int off = 16; off > 0; off >>= 1) v += __shfl_xor(v, off, 32);
  return v;
}

// ---------------------------------------------------------------------------
// fp32 WMMA GEMM:  Out[m,n] = epi( sum_k A[m,k] * W[n,k] )
// Block = 256 threads = 8 waves; C tile 64x64; K chunks of 32 double-buffered
// in LDS via GLOBAL_LOAD_ASYNC_TO_LDS_B128 (ASYNCcnt pipelined).
// LOAD CONTRACT (async path): rows m0..m0+63 of A, n0..n0+63 of W and columns
// 0..32*ceil(K/32)-1 must be readable; columns in [K, 32*ceil(K/32)) must be 0
// on at least the W side (we zero-pad every weight image). Stores are guarded
// by the true M/N.
// EPI==0 : + bias[n], optional relu, row-major store with ldo.
// EPI==1 : cosine-similarity epilogue:
//          m=(b,d), d in [0,MD); n=(a,c), c in [0,NC);
//          Out[((b*(N/NC)+a)*MD+d)*NC+c] = dot / max(fn[m]*sn[n], 1e-8)
// ---------------------------------------------------------------------------
template <int EPI>
__global__ __launch_bounds__(256) void gemm_wmma_kernel(
    const float* __restrict__ A, int lda,
    const float* __restrict__ W, int ldw,
    const float* __restrict__ bias, int relu,
    float* __restrict__ Out, int ldo,
    int M, int N, int K,
    const float* __restrict__ fn, const float* __restrict__ snv,
    int MD, int NC) {
  __shared__ __align__(16) float As[2][64][36];
  __shared__ __align__(16) float Ws[2][64][36];

  const int tid = threadIdx.x;
  const int wave = tid >> 5, lane = tid & 31;
  const int lrow = lane & 15;
  const int khalf = (lane >> 4) << 1;  // lanes 16..31 hold K+2..K+3
  const int m0 = blockIdx.y * 64, n0 = blockIdx.x * 64;
  const int sm0 = wave >> 2, sn0 = wave & 3;
  const int sm1 = sm0 + 2;
  const int nchunks = (K + 31) >> 5;

  v8f acc0 = {};
  v8f acc1 = {};

#if USE_ASYNC_LDS
  // Each wave moves 4 x (32 lanes x 16B) = 2KB per chunk: 2 chunks of the
  // 8KB A tile + 2 of the W tile. 16B unit j of a tile: row = j>>3, 16B col
  // = j&7 (rows are 32 floats = 8 units; LDS row stride 36 floats = 144B,
  // 16B aligned).
  const int u0 = (wave * 2 + 0) * 32 + lane;
  const int u1 = (wave * 2 + 1) * 32 + lane;
  const int rA0 = u0 >> 3, cA0 = (u0 & 7) << 2;
  const int rA1 = u1 >> 3, cA1 = (u1 & 7) << 2;
  const float* gA0 = A + (size_t)(m0 + rA0) * lda + cA0;
  const float* gA1 = A + (size_t)(m0 + rA1) * lda + cA1;
  const float* gW0 = W + (size_t)(n0 + rA0) * ldw + cA0;
  const float* gW1 = W + (size_t)(n0 + rA1) * ldw + cA1;

  // prologue: chunk 0 into buffer 0
  async_copy16(gA0, &As[0][rA0][cA0]);
  async_copy16(gA1, &As[0][rA1][cA1]);
  async_copy16(gW0, &Ws[0][rA0][cA0]);
  async_copy16(gW1, &Ws[0][rA1][cA1]);
#endif

  for (int c = 0; c < nchunks; ++c) {
    const int buf = c & 1;
#if USE_ASYNC_LDS
    if (c + 1 < nchunks) {
      const int kn = (c + 1) << 5;
      async_copy16(gA0 + kn, &As[buf ^ 1][rA0][cA0]);
      async_copy16(gA1 + kn, &As[buf ^ 1][rA1][cA1]);
      async_copy16(gW0 + kn, &Ws[buf ^ 1][rA0][cA0]);
      async_copy16(gW1 + kn, &Ws[buf ^ 1][rA1][cA1]);
      wait_asynccnt<4>();  // in-order: chunk c's 4 transfers are done
    } else {
      wait_asynccnt<0>();
    }
    __syncthreads();
#else
    const int k0 = c << 5;
    for (int it = 0; it < 8; ++it) {
      int i = tid + it * 256;
      int r = i >> 5, cc = i & 31;
      int m = m0 + r, k = k0 + cc, nn = n0 + r;
      As[buf][r][cc] = (m < M && k < K) ? A[(size_t)m * lda + k] : 0.0f;
      Ws[buf][r][cc] = (nn < N && k < K) ? W[(size_t)nn * ldw + k] : 0.0f;
    }
    __syncthreads();
#endif
#pragma unroll
    for (int kk = 0; kk < 8; ++kk) {
      int kb = (kk << 2) + khalf;
      v2f a0, a1, b0;
      a0.x = As[buf][sm0 * 16 + lrow][kb];
      a0.y = As[buf][sm0 * 16 + lrow][kb + 1];
      b0.x = Ws[buf][sn0 * 16 + lrow][kb];
      b0.y = Ws[buf][sn0 * 16 + lrow][kb + 1];
      acc0 = WMMA_F32_K4(a0, b0, acc0);
      a1.x = As[buf][sm1 * 16 + lrow][kb];
      a1.y = As[buf][sm1 * 16 + lrow][kb + 1];
      acc1 = WMMA_F32_K4(a1, b0, acc1);
    }
    __syncthreads();  // buffer free for the next-but-one fill
  }

  // Epilogue. C/D layout: lane<16 -> rows sm*16+0..7, col = sn*16+lane;
  //                       lane>=16 -> rows sm*16+8..15, col = sn*16+lane-16.
  for (int rep = 0; rep < 2; ++rep) {
    v8f acc = rep ? acc1 : acc0;
    int sm = rep ? sm1 : sm0;
    int n = n0 + sn0 * 16 + lrow;
    int mb = m0 + sm * 16 + ((lane >> 4) << 3);
    if (n < N) {
      if (EPI == 0) {
        float bv = bias ? bias[n] : 0.0f;
#pragma unroll
        for (int r = 0; r < 8; ++r) {
          int m = mb + r;
          if (m < M) {
            float v = acc[r] + bv;
            if (relu) v = fmaxf(v, 0.0f);
            Out[(size_t)m * ldo + n] = v;
          }
        }
      } else {
        int AA = N / NC;
        int a_ = n / NC, c_ = n % NC;
        float snx = snv[n];
#pragma unroll
        for (int r = 0; r < 8; ++r) {
          int m = mb + r;
          if (m < M) {
            int b_ = m / MD, d_ = m % MD;
            float den = fmaxf(fn[m] * snx, 1e-8f);
            Out[(((size_t)b_ * AA + a_) * MD + d_) * (size_t)NC + c_] =
                acc[r] / den;
          }
        }
      }
    }
  }
}

// ---------------------------------------------------------------------------
// Attention for one (n, h): scores = QK^T/sqrt(29) (64x64), softmax over t,
// O = P @ V (64x29). dh=29 padded to 32. All matmuls on WMMA fp32.
// qkv layout: [s=0..63][n=0..511][384] with q at +0, k at +116, v at +232.
// obuf layout: [s][n][128], head h occupies cols h*29..h*29+28.
// ---------------------------------------------------------------------------
__global__ __launch_bounds__(256) void attention_kernel(
    const float* __restrict__ qkv, float* __restrict__ obuf) {
  const int n = blockIdx.x >> 2;
  const int h = blockIdx.x & 3;
  const int DH = 29;

  __shared__ float Qs[64][36];
  __shared__ float Ks[64][36];
  __shared__ float Vt[32][68];  // transposed: Vt[d][t]
  __shared__ float Ss[64][68];

  const int tid = threadIdx.x;
  const int wave = tid >> 5, lane = tid & 31;
  const int lrow = lane & 15;
  const int khalf = (lane >> 4) << 1;

  for (int it = 0; it < 8; ++it) {
    int i = tid + it * 256;
    int s = i >> 5, d = i & 31;
    size_t base = ((size_t)s * 512 + n) * 384;
    bool ok = (d < DH);
    Qs[s][d] = ok ? qkv[base + h * DH + d] : 0.0f;
    Ks[s][d] = ok ? qkv[base + 116 + h * DH + d] : 0.0f;
    Vt[d][s] = ok ? qkv[base + 232 + h * DH + d] : 0.0f;
  }
  __syncthreads();

  // scores: M=64 (s), N=64 (t), K=32 (d padded). 16 subtiles, 2 per wave.
  const float scale = 0.18569533817705186f;  // 1/sqrt(29)
  for (int rep = 0; rep < 2; ++rep) {
    int sub = wave + rep * 8;
    int sm = sub >> 2, sn = sub & 3;
    v8f acc = {};
#pragma unroll
    for (int kk = 0; kk < 8; ++kk) {
      int kb = (kk << 2) + khalf;
      v2f a, b;
      a.x = Qs[sm * 16 + lrow][kb];
      a.y = Qs[sm * 16 + lrow][kb + 1];
      b.x = Ks[sn * 16 + lrow][kb];
      b.y = Ks[sn * 16 + lrow][kb + 1];
      acc = WMMA_F32_K4(a, b, acc);
    }
    int rbase = sm * 16 + ((lane >> 4) << 3);
    int col = sn * 16 + lrow;
#pragma unroll
    for (int r = 0; r < 8; ++r) Ss[rbase + r][col] = acc[r] * scale;
  }
  __syncthreads();

  // softmax over t for each row s (threads 0..63, one row each)
  if (tid < 64) {
    float m = -1e30f;
    for (int t = 0; t < 64; ++t) m = fmaxf(m, Ss[tid][t]);
    float s = 0.0f;
    for (int t = 0; t < 64; ++t) {
      float e = expf(Ss[tid][t] - m);
      Ss[tid][t] = e;
      s += e;
    }
    float inv = 1.0f / s;
    for (int t = 0; t < 64; ++t) Ss[tid][t] *= inv;
  }
  __syncthreads();

  // O = P @ V : M=64 (s), N=32 (d), K=64 (t). 8 subtiles, one per wave.
  {
    int sm = wave >> 1, sn = wave & 1;
    v8f acc = {};
#pragma unroll
    for (int kk = 0; kk < 16; ++kk) {
      int kb = (kk << 2) + khalf;
      v2f a, b;
      a.x = Ss[sm * 16 + lrow][kb];
      a.y = Ss[sm * 16 + lrow][kb + 1];
      b.x = Vt[sn * 16 + lrow][kb];
      b.y = Vt[sn * 16 + lrow][kb + 1];
      acc = WMMA_F32_K4(a, b, acc);
    }
    int rbase = sm * 16 + ((lane >> 4) << 3);
    int d = sn * 16 + lrow;
    if (d < DH) {
#pragma unroll
      for (int r = 0; r < 8; ++r) {
        int s = rbase + r;
        obuf[((size_t)s * 512 + n) * 128 + h * DH + d] = acc[r];
      }
    }
  }
}

// ---------------------------------------------------------------------------
// y = LayerNorm(a + b) * g + beta (row length E_, row stride ld, 1 wave/row)
// ---------------------------------------------------------------------------
__global__ __launch_bounds__(256) void add_ln_kernel(
    const float* __restrict__ a, const float* __restrict__ b,
    const float* __restrict__ g, const float* __restrict__ beta,
    float* __restrict__ out, int M, int E_, int ld) {
  int row = blockIdx.x * (blockDim.x >> 5) + (threadIdx.x >> 5);
  int lane = threadIdx.x & 31;
  if (row >= M) return;
  const float* pa = a + (size_t)row * ld;
  const float* pb = b + (size_t)row * ld;
  float s = 0.0f, s2 = 0.0f;
  for (int i = lane; i < E_; i += 32) {
    float v = pa[i] + pb[i];
    s += v;
    s2 += v * v;
  }
  s = wave_sum(s);
  s2 = wave_sum(s2);
  float mean = s / (float)E_;
  float var = s2 / (float)E_ - mean * mean;
  float inv = rsqrtf(var + 1e-5f);
  float* po = out + (size_t)row * ld;
  for (int i = lane; i < E_; i += 32) {
    float v = pa[i] + pb[i];
    po[i] = (v - mean) * inv * g[i] + beta[i];
  }
}

// ---------------------------------------------------------------------------
// Row L2 norms: out[row] = max(sqrt(sum_k A[row,k]^2), floorv). One wave/row.
// ---------------------------------------------------------------------------
__global__ __launch_bounds__(256) void rownorm_kernel(
    const float* __restrict__ A, int ld, int K, int M, float floorv,
    float* __restrict__ out) {
  int row = blockIdx.x * (blockDim.x >> 5) + (threadIdx.x >> 5);
  int lane = threadIdx.x & 31;
  if (row >= M) return;
  const float* p = A + (size_t)row * ld;
  float s = 0.0f;
  for (int i = lane; i < K; i += 32) {
    float v = p[i];
    s += v * v;
  }
  s = wave_sum(s);
  if (lane == 0) out[row] = fmaxf(sqrtf(s), floorv);
}

// ---------------------------------------------------------------------------
// Zero-padded image copy: dst (Rpad x S) <- src (R x C), tails = 0.
// ---------------------------------------------------------------------------
__global__ void pad_copy_kernel(const float* __restrict__ src, int R, int C,
                                int S, int Rpad, float* __restrict__ dst) {
  int id = blockIdx.x * 256 + threadIdx.x;
  int total = Rpad * S;
  if (id >= total) return;
  int r = id / S, c = id % S;
  dst[id] = (r < R && c < C) ? src[(size_t)r * C + c] : 0.0f;
}

// ---------------------------------------------------------------------------
// Small permutes
// ---------------------------------------------------------------------------
// outsT[(b*116+e)*512+t] = out1[(b*512+t)*128+e]   (out1 row stride 128)
__global__ void transpose_out1_kernel(const float* __restrict__ in,
                                      float* __restrict__ out) {
  int id = blockIdx.x * 256 + threadIdx.x;
  const int TOT = 64 * 116 * 512;
  if (id >= TOT) return;
  int t = id % 512;
  int r = id / 512;
  int e = r % 116;
  int b = r / 116;
  out[id] = in[((size_t)b * 512 + t) * 128 + e];
}

// fea[(b*64+h)*116+d] = fea_[(b*116+d)*64+h]
__global__ void transpose_fea_kernel(const float* __restrict__ in,
                                     float* __restrict__ out) {
  int id = blockIdx.x * 256 + threadIdx.x;
  const int TOT = 64 * 64 * 116;
  if (id >= TOT) return;
  int d = id % 116;
  int r = id / 116;
  int hh = r % 64;
  int b = r / 64;
  out[id] = in[((size_t)b * 116 + d) * 64 + hh];
}

// simW[(a*90+c)*64+h] = sim[(a*64+h)*90+c]
__global__ void transpose_sim_kernel(const float* __restrict__ in,
                                     float* __restrict__ out) {
  int id = blockIdx.x * 256 + threadIdx.x;
  const int TOT = 116 * 90 * 64;
  if (id >= TOT) return;
  int hh = id % 64;
  int r = id / 64;
  int c = r % 90;
  int a = r / 90;
  out[id] = in[((size_t)a * 64 + hh) * 90 + c];
}

// similar[id] = similarity[id] / nrm[id / rowlen]
__global__ void scale_rows_kernel(const float* __restrict__ in,
                                  const float* __restrict__ nrm,
                                  float* __restrict__ out,
                                  unsigned int total, int rowlen) {
  unsigned int id = blockIdx.x * 256u + threadIdx.x;
  if (id >= total) return;
  out[id] = in[id] / nrm[id / (unsigned int)rowlen];
}

// ---------------------------------------------------------------------------
// Host orchestration
// ---------------------------------------------------------------------------
extern "C" void kernel_launch(void* const* d_in, const int* in_sizes, int n_in,
                              void* d_out, int out_size, void* d_ws,
                              size_t ws_size, hipStream_t stream) {
  (void)in_sizes; (void)n_in; (void)out_size; (void)ws_size;

  const float* x = (const float*)d_in[0];
  const float* sim = (const float*)d_in[1];
  const float* P[2][12];
  for (int L = 0; L < 2; ++L)
    for (int j = 0; j < 12; ++j) P[L][j] = (const float*)d_in[2 + L * 12 + j];
  const float* fc_w = (const float*)d_in[26];
  const float* fc_b = (const float*)d_in[27];

  // Padded workspace layout (floats)
  float* ws = (float*)d_ws;
  float* ws_x = ws;                          // 32768 x 128
  float* ws_qkv = ws_x + 32768ull * 128;     // 32768 x 384 (also hidden/outsT)
  float* ws_o = ws_qkv + 32768ull * 384;     // 32768 x 128
  float* ws_t4 = ws_o + 32768ull * 128;      // 32768 x 128
  float* ws_y = ws_t4 + 32768ull * 128;      // 32768 x 128
  float* ws_fea = ws_y + 32768ull * 128;     // 7424 x 64
  float* ws_simw = ws_fea + 7424ull * 64;    // 10496 x 64 (rows >=10440 unused)
  float* ws_fn = ws_simw + 10496ull * 64;    // 7424
  float* ws_sn = ws_fn + 7424;               // 10440
  float* ws_nrm = ws_sn + 10440;             // 7424
  float* ws_wpad = ws_nrm + 7424;            // padded weight images
  float* wqkv_p[2], *wo_p[2], *w1_p[2], *w2_p[2];
  {
    float* p = ws_wpad;
    for (int L = 0; L < 2; ++L) {
      wqkv_p[L] = p; p += 384ull * 128;    // 384 x 128
      wo_p[L] = p;   p += 128ull * 128;    // 128 x 128
      w1_p[L] = p;   p += 2048ull * 128;   // 2048 x 128
      w2_p[L] = p;   p += 128ull * 2048;   // 128 x 2048
    }
  }

  dim3 blk(256);
#define PAD_COPY(src, R, C, S, Rpad, dst) \
  pad_copy_kernel<<<((Rpad) * (S) + 255) / 256, blk, 0, stream>>>(src, R, C, S, Rpad, dst)

  // Stage zero-padded operand images (async-load contract: in-range, 0 tails)
  PAD_COPY(x, 32768, 116, 128, 32768, ws_x);
  for (int L = 0; L < 2; ++L) {
    PAD_COPY(P[L][0], 348, 116, 128, 384, wqkv_p[L]);
    PAD_COPY(P[L][2], 116, 116, 128, 128, wo_p[L]);
    PAD_COPY(P[L][6], 2048, 116, 128, 2048, w1_p[L]);
    PAD_COPY(P[L][8], 116, 2048, 2048, 128, w2_p[L]);
  }

  for (int L = 0; L < 2; ++L) {
    const float *bqkv = P[L][1], *bo = P[L][3];
    const float *g1 = P[L][4], *bn1 = P[L][5], *b1 = P[L][7];
    const float *b2 = P[L][9], *g2 = P[L][10], *bn2 = P[L][11];

    // qkv = x @ wqkv.T + bqkv : (32768 x 116) x (348 x 116)^T
    gemm_wmma_kernel<0><<<dim3(6, 512), blk, 0, stream>>>(
        ws_x, 128, wqkv_p[L], 128, bqkv, 0, ws_qkv, 384, 32768, 348, 116,
        nullptr, nullptr, 0, 0);
    // attention per (n, h)
    attention_kernel<<<512 * 4, blk, 0, stream>>>(ws_qkv, ws_o);
    // o = o @ wo.T + bo
    gemm_wmma_kernel<0><<<dim3(2, 512), blk, 0, stream>>>(
        ws_o, 128, wo_p[L], 128, bo, 0, ws_t4, 128, 32768, 116, 116,
        nullptr, nullptr, 0, 0);
    // y = LN(x + o)
    add_ln_kernel<<<4096, blk, 0, stream>>>(ws_x, ws_t4, g1, bn1, ws_y,
                                            32768, 116, 128);
    // FFN in 8 chunks of 4096 rows (hidden reuses the qkv buffer)
    for (int c = 0; c < 8; ++c) {
      size_t r0 = (size_t)c * 4096;
      gemm_wmma_kernel<0><<<dim3(32, 64), blk, 0, stream>>>(
          ws_y + r0 * 128, 128, w1_p[L], 128, b1, 1, ws_qkv, 2048,
          4096, 2048, 116, nullptr, nullptr, 0, 0);
      gemm_wmma_kernel<0><<<dim3(2, 64), blk, 0, stream>>>(
          ws_qkv, 2048, w2_p[L], 2048, b2, 0, ws_t4 + r0 * 128, 128,
          4096, 116, 2048, nullptr, nullptr, 0, 0);
    }
    // x = LN(y + f)
    add_ln_kernel<<<4096, blk, 0, stream>>>(ws_y, ws_t4, g2, bn2, ws_x,
                                            32768, 116, 128);
  }

  // outsT[b*116+e][t] = out1[b][t][e]
  transpose_out1_kernel<<<(3801088 + 255) / 256, blk, 0, stream>>>(ws_x,
                                                                   ws_qkv);
  // fea_ = outsT @ fc_w.T + fc_b : (7424 x 512) x (64 x 512)^T (exact tiles)
  gemm_wmma_kernel<0><<<dim3(1, 116), blk, 0, stream>>>(
      ws_qkv, 512, fc_w, 512, fc_b, 0, ws_fea, 64, 7424, 64, 512,
      nullptr, nullptr, 0, 0);

  float* out_f = (float*)d_out;
  float* out_fea = out_f;                    // 475,136
  float* out_simy = out_f + 475136;          // 77,506,560
  float* out_simr = out_simy + 77506560ull;  // 77,506,560

  transpose_fea_kernel<<<(475136 + 255) / 256, blk, 0, stream>>>(ws_fea,
                                                                 out_fea);
  transpose_sim_kernel<<<(668160 + 255) / 256, blk, 0, stream>>>(sim, ws_simw);

  rownorm_kernel<<<(7424 + 7) / 8, blk, 0, stream>>>(ws_fea, 64, 64, 7424,
                                                     0.0f, ws_fn);
  rownorm_kernel<<<(10440 + 7) / 8, blk, 0, stream>>>(ws_simw, 64, 64, 10440,
                                                      0.0f, ws_sn);

  // similarity = (fea_ @ simW^T) / max(fn*sn, 1e-8), scattered to [b][a][d][c]
  gemm_wmma_kernel<1><<<dim3(164, 116), blk, 0, stream>>>(
      ws_fea, 64, ws_simw, 64, nullptr, 0, out_simy, 0, 7424, 10440, 64,
      ws_fn, ws_sn, 116, 90);

  // similar = similarity / max(||row||, 1e-12), rows of length 10440
  rownorm_kernel<<<(7424 + 7) / 8, blk, 0, stream>>>(out_simy, 10440, 10440,
                                                     7424, 1e-12f, ws_nrm);
  scale_rows_kernel<<<(77506560u + 255u) / 256u, blk, 0, stream>>>(
      out_simy, ws_nrm, out_simr, 77506560u, 10440);
}